// SpatialTransformerNetwork_74277164417317
// MI455X (gfx1250) — compile-verified
//
#include <hip/hip_runtime.h>
#include <hip/hip_bf16.h>

// TPS spatial transformer for MI455X (gfx1250, wave32).
//  Kernel 1: per-batch 18x18 Gaussian-elimination solve (1 wave / batch).
//  Kernel 2: warp+sample. Each wave owns 32 pixels = two 16x16 WMMA tiles;
//            the phi(r2)(16x20pad) x nodes(20padx2) product is accumulated with
//            V_WMMA_F32_16X16X4_F32 (f32 WMMA: phi can reach ~2.6e6, which
//            overflows f16, so the f32 matrix path is the correct precision).
//            phi uses the raw v_log_f32 (log2) + ln2/2 scale, branchless:
//            this kernel is transcendental-bound, so minimizing per-phi VALU
//            work and avoiding exec-divergence is the dominant optimization.
//            wx/wy are redistributed lane-wise via per-wave LDS + s_wait_dscnt.

typedef __attribute__((ext_vector_type(2))) float v2f;
typedef __attribute__((ext_vector_type(8))) float v8f;

#define TPS_B 16
#define TPS_C 3
#define TPS_H 512
#define TPS_W 384
#define TPS_N 18
#define TPS_NP 20                 // N padded to a multiple of 4 (K-chunks)
#define TPS_HW (TPS_H * TPS_W)
#define TPS_EPS 1e-5f

__device__ __forceinline__ float tps_phi(float dx, float dy) {
    float r2 = dx * dx + dy * dy;
    // 0.5*r2*ln(r2) = (ln2/2) * r2 * log2(r2).
    // fmax clamp: at r2==0 the log argument is 1e-35 (finite), and
    // r2 * log2(...) == 0 exactly -> phi(0) = 0 with no compare/select.
    float l2 = __builtin_amdgcn_logf(fmaxf(r2, 1e-35f));   // raw v_log_f32
    return 0.34657359027997264f * r2 * l2;                 // ln(2)/2
}

// ---------------------------------------------------------------------------
// Kernel 1: per-batch TPS solve.  grid = B blocks, 32 threads (one wave32).
// Builds Aug[18][20] = [ phi(d2) - eps*I | control_points ] in LDS and does
// Gaussian elimination with partial pivoting; back-substitutes 2 RHS columns.
// ---------------------------------------------------------------------------
__global__ void __launch_bounds__(32) tps_solve_kernel(
    const float* __restrict__ kp,      // (B, 18, 2)
    const float* __restrict__ cp,      // (18, 2)
    float* __restrict__ nodes)         // (B, 18, 2) out
{
    __shared__ float Aug[TPS_N][TPS_N + 2];
    __shared__ float dstS[2 * TPS_N];
    __shared__ int piv;

    const int b = blockIdx.x;
    const int tid = threadIdx.x;

    for (int i = tid; i < 2 * TPS_N; i += 32) dstS[i] = kp[b * 2 * TPS_N + i];
    __syncthreads();

    for (int idx = tid; idx < TPS_N * (TPS_N + 2); idx += 32) {
        int i = idx / (TPS_N + 2);
        int j = idx % (TPS_N + 2);
        float v;
        if (j < TPS_N) {
            float dx = dstS[i * 2]     - dstS[j * 2];
            float dy = dstS[i * 2 + 1] - dstS[j * 2 + 1];
            v = tps_phi(dx, dy) - (i == j ? TPS_EPS : 0.0f);
        } else {
            v = cp[i * 2 + (j - TPS_N)];
        }
        Aug[i][j] = v;
    }
    __syncthreads();

    for (int k = 0; k < TPS_N; ++k) {
        if (tid == 0) {
            int p = k;
            float best = fabsf(Aug[k][k]);
            for (int r = k + 1; r < TPS_N; ++r) {
                float v = fabsf(Aug[r][k]);
                if (v > best) { best = v; p = r; }
            }
            piv = p;
        }
        __syncthreads();
        int p = piv;
        if (p != k && tid < TPS_N + 2) {
            float t = Aug[k][tid]; Aug[k][tid] = Aug[p][tid]; Aug[p][tid] = t;
        }
        __syncthreads();
        if (tid > k && tid < TPS_N) {
            float f = Aug[tid][k] / Aug[k][k];
            for (int j = k; j < TPS_N + 2; ++j) Aug[tid][j] -= f * Aug[k][j];
        }
        __syncthreads();
    }

    // back substitution: thread 0 -> x column, thread 1 -> y column
    if (tid < 2) {
        const int c = TPS_N + tid;
        float x[TPS_N];
        for (int i = TPS_N - 1; i >= 0; --i) {
            float s = Aug[i][c];
            for (int j = i + 1; j < TPS_N; ++j) s -= Aug[i][j] * x[j];
            x[i] = s / Aug[i][i];
        }
        for (int i = 0; i < TPS_N; ++i) nodes[b * 2 * TPS_N + i * 2 + tid] = x[i];
    }
}

// ---------------------------------------------------------------------------
// Kernel 2: warp + bilinear sample.
// block = 256 threads (8 waves); wave -> 32 pixels (two 16x16 WMMA tiles).
// grid = (HW/256, B).
// dst/nodes are padded to 20 keypoints with zeros: padded B-matrix rows are
// zero, so the (unconditionally computed) padded A entries contribute nothing.
// ---------------------------------------------------------------------------
__global__ void __launch_bounds__(256) tps_warp_kernel(
    const float* __restrict__ cloth,   // (B, 3, H, W)
    const float* __restrict__ kp,      // (B, 18, 2)  dst keypoints
    const float* __restrict__ nodes,   // (B, 18, 2)  solved TPS nodes
    float* __restrict__ out)           // (B, 3, H, W)
{
    __shared__ float s_dst[2 * TPS_NP];   // padded with zeros
    __shared__ float s_nod[2 * TPS_NP];   // padded with zeros (zero B rows)
    __shared__ float s_wxy[8][32][2];

    const int b   = blockIdx.y;
    const int tid = threadIdx.x;

    if (tid < 2 * TPS_NP) {
        s_dst[tid] = (tid < 2 * TPS_N) ? kp[b * 2 * TPS_N + tid] : 0.0f;
    } else if (tid < 4 * TPS_NP) {
        int i = tid - 2 * TPS_NP;
        s_nod[i] = (i < 2 * TPS_N) ? nodes[b * 2 * TPS_N + i] : 0.0f;
    }
    __syncthreads();

    const int lane = tid & 31;
    const int wave = tid >> 5;
    const int half = lane >> 4;   // selects K pair {0,1} vs {2,3} of the chunk
    const int row  = lane & 15;   // A-matrix M row / D-matrix N column
    const int pixBase = blockIdx.x * 256 + wave * 32;

    // pixel coords of this lane's A rows for both 16-pixel tiles
    float gx[2], gy[2];
#pragma unroll
    for (int t = 0; t < 2; ++t) {
        int p = pixBase + t * 16 + row;
        gx[t] = (float)(p % TPS_W);
        gy[t] = (float)(p / TPS_W);
    }

    v8f acc0 = {};   // D tile for pixels [pixBase .. +15]
    v8f acc1 = {};   // D tile for pixels [pixBase+16 .. +31]

    // 5 K-chunks of 4 keypoints (N padded 18 -> 20)
#pragma unroll
    for (int chunk = 0; chunk < 5; ++chunk) {
        const int n0 = chunk * 4 + half * 2;

        // B operand (4x16): row K holds nodes; col 0 = node.x, col 1 = node.y
        v2f bv = {0.0f, 0.0f};
        if (row < 2) {
            bv.x = s_nod[n0 * 2 + row];
            bv.y = s_nod[n0 * 2 + 2 + row];
        }

        const float d0x = s_dst[n0 * 2];
        const float d0y = s_dst[n0 * 2 + 1];
        const float d1x = s_dst[n0 * 2 + 2];
        const float d1y = s_dst[n0 * 2 + 3];

        // A operand (16x4 f32): lane = row M, two K values per lane
        v2f a0 = { tps_phi(gx[0] - d0x, gy[0] - d0y),
                   tps_phi(gx[0] - d1x, gy[0] - d1y) };
        v2f a1 = { tps_phi(gx[1] - d0x, gy[1] - d0y),
                   tps_phi(gx[1] - d1x, gy[1] - d1y) };

        acc0 = __builtin_amdgcn_wmma_f32_16x16x4_f32(false, a0, false, bv,
                                                     (short)0, acc0, false, false);
        acc1 = __builtin_amdgcn_wmma_f32_16x16x4_f32(false, a1, false, bv,
                                                     (short)0, acc1, false, false);
    }

    // D layout: lanes 0/16 hold column N=0 (wx), lanes 1/17 hold N=1 (wy);
    // VGPR r holds M=r (lanes 0-15) or M=8+r (lanes 16-31).
    if (row < 2) {
#pragma unroll
        for (int r = 0; r < 8; ++r) {
            s_wxy[wave][half * 8 + r][row]      = acc0[r];
            s_wxy[wave][16 + half * 8 + r][row] = acc1[r];
        }
    }
    asm volatile("s_wait_dscnt 0" ::: "memory");   // per-wave LDS RAW fence

    float wx = s_wxy[wave][lane][0];
    float wy = s_wxy[wave][lane][1];

    // clip exactly like the reference
    wx = fminf(fmaxf(wx, 0.0f), (float)(TPS_W - 1));
    wy = fminf(fmaxf(wy, 0.0f), (float)(TPS_H - 1));

    float xf = floorf(wx), yf = floorf(wy);
    int x0 = (int)xf, y0 = (int)yf;
    int x1 = x0 + 1,  y1 = y0 + 1;
    x0 = min(max(x0, 0), TPS_W - 1);  x1 = min(max(x1, 0), TPS_W - 1);
    y0 = min(max(y0, 0), TPS_H - 1);  y1 = min(max(y1, 0), TPS_H - 1);

    float fx0 = (float)x0, fx1 = (float)x1, fy0 = (float)y0, fy1 = (float)y1;
    float wa = (fx1 - wx) * (fy1 - wy);
    float wb = (fx1 - wx) * (wy - fy0);
    float wc = (wx - fx0) * (fy1 - wy);
    float wd = (wx - fx0) * (wy - fy0);

    const int p = pixBase + lane;
    const float* img = cloth + (size_t)b * TPS_C * TPS_HW;
#pragma unroll
    for (int c = 0; c < TPS_C; ++c) {
        const float* ch = img + (size_t)c * TPS_HW;
        float Ia = ch[y0 * TPS_W + x0];
        float Ib = ch[y1 * TPS_W + x0];
        float Ic = ch[y0 * TPS_W + x1];
        float Id = ch[y1 * TPS_W + x1];
        out[((size_t)b * TPS_C + c) * TPS_HW + p] = wa * Ia + wb * Ib + wc * Ic + wd * Id;
    }
}

extern "C" void kernel_launch(void* const* d_in, const int* in_sizes, int n_in,
                              void* d_out, int out_size, void* d_ws, size_t ws_size,
                              hipStream_t stream) {
    (void)in_sizes; (void)n_in; (void)out_size; (void)ws_size;
    const float* cloth = (const float*)d_in[0];   // (16,3,512,384)
    const float* kp    = (const float*)d_in[1];   // (16,18,2)
    const float* cp    = (const float*)d_in[2];   // (18,2)
    float* out   = (float*)d_out;                 // (16,3,512,384)
    float* nodes = (float*)d_ws;                  // B*18*2 floats scratch

    tps_solve_kernel<<<TPS_B, 32, 0, stream>>>(kp, cp, nodes);

    dim3 grid(TPS_HW / 256, TPS_B);
    tps_warp_kernel<<<grid, 256, 0, stream>>>(cloth, kp, nodes, out);
}